// GGNN_78383153152127
// MI455X (gfx1250) — compile-verified
//
#include <hip/hip_runtime.h>

// GGNN: B=16, N=1024, H=256, E=16384, 4 layers. All f32.
// GEMMs via V_WMMA_F32_16X16X4_F32 (true-f32 matrix pipe, matches reference numerics).
// Scatter-add via global_atomic_add_f32 into L2 (whole working set is L2-resident).

#define BATCH 16
#define NN    1024
#define HH    256
#define EE    16384
#define RR    (BATCH * NN)   // 16384 rows

typedef __attribute__((ext_vector_type(2))) float v2f;
typedef __attribute__((ext_vector_type(8))) float v8f;

__device__ __forceinline__ v8f wmma_f32(v2f a, v2f b, v8f c) {
  // D = A(16x4 f32) x B(4x16 f32) + C(16x16 f32)
  return __builtin_amdgcn_wmma_f32_16x16x4_f32(
      /*neg_a=*/false, a, /*neg_b=*/false, b,
      /*c_mod=*/(short)0, c, /*reuse_a=*/false, /*reuse_b=*/false);
}

__device__ __forceinline__ float sigmoidf_(float x) {
  return 1.0f / (1.0f + __expf(-x));
}

// ---------------------------------------------------------------------------
// m = x @ W   : X (RR,HH) row-major, W (HH,HH) row-major, M (RR,HH)
// Block = 128 threads (4 waves). Each wave: 16 rows x 64 cols (4 WMMA tiles,
// A-fragment reused 4x per k-step). Grid = RR/16 blocks.
// A 16x4 f32 layout: lane&15 = M-row; lane>>4 selects K pair {0,1} vs {2,3};
// VGPR v within the float2 = K LSB. B mirrored: lane&15 = N-col.
// ---------------------------------------------------------------------------
__global__ void __launch_bounds__(128)
ggnn_gemm_xw(const float* __restrict__ X, const float* __restrict__ W,
             float* __restrict__ M) {
  const int lane = threadIdx.x & 31;
  const int wave = threadIdx.x >> 5;     // 0..3 -> 64-col strip
  const int l16  = lane & 15;
  const int half = lane >> 4;            // 0 or 1
  const int row0 = blockIdx.x * 16;
  const int col0 = wave * 64;

  v8f acc[4] = {};
  const float* xrow = X + (size_t)(row0 + l16) * HH;

  for (int k0 = 0; k0 < HH; k0 += 4) {
    const int ka = k0 + half * 2;
    v2f a = *(const v2f*)(xrow + ka);    // contiguous K pair
#pragma unroll
    for (int j = 0; j < 4; ++j) {
      const int col = col0 + j * 16 + l16;
      v2f b;
      b.x = W[(size_t)ka * HH + col];
      b.y = W[(size_t)(ka + 1) * HH + col];
      acc[j] = wmma_f32(a, b, acc[j]);
    }
  }

#pragma unroll
  for (int j = 0; j < 4; ++j) {
#pragma unroll
    for (int v = 0; v < 8; ++v) {
      const int row = row0 + v + half * 8;       // C/D layout: VGPR v -> M=v / v+8
      const int col = col0 + j * 16 + l16;
      M[(size_t)row * HH + col] = acc[j][v];
    }
  }
}

// ---------------------------------------------------------------------------
// agg = 0
// ---------------------------------------------------------------------------
__global__ void __launch_bounds__(256)
ggnn_zero(float4* __restrict__ p, int n4) {
  const int i = blockIdx.x * 256 + threadIdx.x;
  if (i < n4) p[i] = make_float4(0.f, 0.f, 0.f, 0.f);
}

// ---------------------------------------------------------------------------
// agg[b, dst[e], h] += m[b, src[e], h]   (one block per edge, lane per h)
// ---------------------------------------------------------------------------
__global__ void __launch_bounds__(256)
ggnn_scatter(const float* __restrict__ M, const int* __restrict__ src,
             const int* __restrict__ dst, float* __restrict__ AGG) {
  const int e = blockIdx.x;
  const int h = threadIdx.x;
  const int s = src[e];
  const int d = dst[e];
  const float* mp = M + (size_t)s * HH + h;
  float*       ap = AGG + (size_t)d * HH + h;
#pragma unroll
  for (int b = 0; b < BATCH; ++b) {
    unsafeAtomicAdd(ap + (size_t)b * NN * HH, mp[(size_t)b * NN * HH]);
  }
}

// ---------------------------------------------------------------------------
// Fused GRU cell. Per wave: one 16x16 output tile.
//   gi = agg @ w_ih^T + b_ih ; gh = x @ w_hh^T + b_hh (6 f32-WMMA accumulators)
//   r = sig(i_r+h_r); z = sig(i_z+h_z); n = tanh(i_n + r*h_n)
//   out = (1-z)*n + z*x
// w_ih/w_hh are (3H,H) row-major -> transposed B fragments are contiguous
// float2 loads per lane. Grid (RR/16, HH/64), block 128 (4 waves).
// ---------------------------------------------------------------------------
__global__ void __launch_bounds__(128)
ggnn_gru(const float* __restrict__ AGG, const float* __restrict__ X,
         const float* __restrict__ W_ih, const float* __restrict__ W_hh,
         const float* __restrict__ B_ih, const float* __restrict__ B_hh,
         float* __restrict__ OUT) {
  const int lane = threadIdx.x & 31;
  const int wave = threadIdx.x >> 5;     // 0..3
  const int l16  = lane & 15;
  const int half = lane >> 4;
  const int row0 = blockIdx.x * 16;
  const int colb = blockIdx.y * 64 + wave * 16;  // tile col base
  const int c    = colb + l16;                   // this lane's output column

  v8f a_ir = {}, a_iz = {}, a_in = {};
  v8f a_hr = {}, a_hz = {}, a_hn = {};

  const float* arow = AGG + (size_t)(row0 + l16) * HH;
  const float* xrow = X   + (size_t)(row0 + l16) * HH;
  // B fragment sources: row j of W (= column j of W^T); gate g at rows g*H..
  const float* pir = W_ih + (size_t)(0 * HH + c) * HH;
  const float* piz = W_ih + (size_t)(1 * HH + c) * HH;
  const float* pin = W_ih + (size_t)(2 * HH + c) * HH;
  const float* phr = W_hh + (size_t)(0 * HH + c) * HH;
  const float* phz = W_hh + (size_t)(1 * HH + c) * HH;
  const float* phn = W_hh + (size_t)(2 * HH + c) * HH;

  for (int k0 = 0; k0 < HH; k0 += 4) {
    const int ka = k0 + half * 2;
    v2f aA = *(const v2f*)(arow + ka);
    v2f aX = *(const v2f*)(xrow + ka);
    v2f b;
    b = *(const v2f*)(pir + ka); a_ir = wmma_f32(aA, b, a_ir);
    b = *(const v2f*)(piz + ka); a_iz = wmma_f32(aA, b, a_iz);
    b = *(const v2f*)(pin + ka); a_in = wmma_f32(aA, b, a_in);
    b = *(const v2f*)(phr + ka); a_hr = wmma_f32(aX, b, a_hr);
    b = *(const v2f*)(phz + ka); a_hz = wmma_f32(aX, b, a_hz);
    b = *(const v2f*)(phn + ka); a_hn = wmma_f32(aX, b, a_hn);
  }

  const float bir = B_ih[c],          bhr = B_hh[c];
  const float biz = B_ih[c + HH],     bhz = B_hh[c + HH];
  const float bin = B_ih[c + 2 * HH], bhn = B_hh[c + 2 * HH];

#pragma unroll
  for (int v = 0; v < 8; ++v) {
    const int row = row0 + v + half * 8;
    const float r = sigmoidf_(a_ir[v] + bir + a_hr[v] + bhr);
    const float z = sigmoidf_(a_iz[v] + biz + a_hz[v] + bhz);
    const float n = tanhf(a_in[v] + bin + r * (a_hn[v] + bhn));
    const float hprev = X[(size_t)row * HH + c];
    OUT[(size_t)row * HH + c] = (1.0f - z) * n + z * hprev;
  }
}

// ---------------------------------------------------------------------------
extern "C" void kernel_launch(void* const* d_in, const int* in_sizes, int n_in,
                              void* d_out, int out_size, void* d_ws, size_t ws_size,
                              hipStream_t stream) {
  const float* emb    = (const float*)d_in[0];   // (B,N,H)
  const int*   eidx   = (const int*)d_in[1];     // (2,E) int32
  const float* weight = (const float*)d_in[2];   // (4,H,H)
  const float* w_ih   = (const float*)d_in[3];   // (3H,H)
  const float* w_hh   = (const float*)d_in[4];   // (3H,H)
  const float* b_ih   = (const float*)d_in[5];   // (3H)
  const float* b_hh   = (const float*)d_in[6];   // (3H)
  float* out = (float*)d_out;

  const size_t plane = (size_t)RR * HH;          // 16.8 MB / buffer
  float* bufA = (float*)d_ws;
  float* bufB = bufA + plane;
  float* agg  = bufB + plane;

  const int* src = eidx;
  const int* dst = eidx + EE;

  const float* xin = emb;
  for (int l = 0; l < 4; ++l) {
    float* m    = (l & 1) ? bufB : bufA;          // never aliases xin
    float* xout = (l == 3) ? out : m;             // m is dead by GRU time

    ggnn_gemm_xw<<<RR / 16, 128, 0, stream>>>(xin, weight + (size_t)l * HH * HH, m);
    ggnn_zero<<<(int)(plane / 4 + 255) / 256, 256, 0, stream>>>((float4*)agg,
                                                                (int)(plane / 4));
    ggnn_scatter<<<EE, 256, 0, stream>>>(m, src, dst, agg);
    ggnn_gru<<<dim3(RR / 16, HH / 64), 128, 0, stream>>>(agg, xin, w_ih, w_hh,
                                                         b_ih, b_hh, xout);
    xin = xout;
  }
}